// Matching_3934190044174
// MI455X (gfx1250) — compile-verified
//
#include <hip/hip_runtime.h>
#include <stdint.h>

// Problem constants (match reference)
#define DD   128
#define HH   160
#define WW   160
#define CCH  12
#define NKPS 512
#define LW   9            // 9 displacements per axis (-8..8 step 2)
#define PW   4            // 4 patch offsets per axis (-3..3 step 2)
#define WIN  12           // 12 window points per axis (parity grid of p+d)
#define NDISP (LW*LW*LW)  // 729
#define COST_ELEMS (NKPS*NDISP)
#define HWXY (HH*WW)
#define DHW  ((size_t)DD*HH*WW)

#define NTHREADS 192      // 6 wave32 = 6 N-tiles of 16 columns

// ---- gfx1250 feature detection ----
#if defined(__has_builtin)
#if __has_builtin(__builtin_amdgcn_global_load_async_to_lds_b32)
#define HAVE_ASYNC_LDS 1
#endif
#if __has_builtin(__builtin_amdgcn_wmma_f32_16x16x4_f32)
#define HAVE_WMMA_F32 1
#endif
#endif

typedef __attribute__((address_space(1))) int g_int_t;   // global
typedef __attribute__((address_space(3))) int l_int_t;   // LDS
typedef float v2f __attribute__((ext_vector_type(2)));
typedef float v8f __attribute__((ext_vector_type(8)));

#if defined(HAVE_ASYNC_LDS)
__device__ __forceinline__ void async_copy_b32(const float* gsrc, float* lds_dst) {
  float* g = const_cast<float*>(gsrc);
  __builtin_amdgcn_global_load_async_to_lds_b32(
      (g_int_t*)g, (l_int_t*)lds_dst, /*offset=*/0, /*cpol=*/0);
}
__device__ __forceinline__ void wait_async_zero() {
#if __has_builtin(__builtin_amdgcn_s_wait_asynccnt)
  __builtin_amdgcn_s_wait_asynccnt(0);
#else
  asm volatile("s_wait_asynccnt 0" ::: "memory");
#endif
}
#endif

#if defined(HAVE_WMMA_F32)
__device__ __forceinline__ v8f wmma_f32_4(v2f a, v2f b, v8f c) {
  // (neg_a, A, neg_b, B, c_mod, C, reuse_a, reuse_b)
  return __builtin_amdgcn_wmma_f32_16x16x4_f32(false, a, false, b, (short)0, c,
                                               false, false);
}
#endif

__device__ __forceinline__ int clampi(int v, int lo, int hi) {
  return v < lo ? lo : (v > hi ? hi : v);
}

// One block per keypoint. 192 threads (6 wave32).
__global__ __launch_bounds__(NTHREADS) void match_cost_kernel(
    const int* __restrict__ kps,      // (NKPS,3) z,y,x
    const float* __restrict__ ffix,   // (C,D,H,W)
    const float* __restrict__ fmov,   // (C,D,H,W)
    float* __restrict__ out)          // cost (NKPS*729) then disp (729*3)
{
  __shared__ __align__(16) float s_mov[CCH][WIN][WIN][WIN]; // 82944 B
  __shared__ __align__(16) float s_pf [CCH][PW][PW][PW];    //  3072 B (holds -2*pf in WMMA path)
#if defined(HAVE_WMMA_F32)
  __shared__ float s_sff;                                   // 4*sum(pf^2)
#endif

  const int k   = blockIdx.x;
  const int tid = threadIdx.x;

  const int kz = kps[3*k + 0];
  const int ky = kps[3*k + 1];
  const int kx = kps[3*k + 2];

#if defined(HAVE_WMMA_F32)
  if (tid == 0) s_sff = 0.0f;
#endif

  // ---- Stage moving window (async global->LDS): coords kz-11+2*zw etc. ----
  const int bz = kz - 11, by = ky - 11, bx = kx - 11;
  #pragma unroll 1
  for (int l = 0; l < (CCH*WIN*WIN*WIN)/NTHREADS; ++l) {   // 108 iters
    int i  = tid + l*NTHREADS;
    int xw = i % WIN;
    int t1 = i / WIN;
    int yw = t1 % WIN;
    int t2 = t1 / WIN;
    int zw = t2 % WIN;
    int c  = t2 / WIN;
    int gz = clampi(bz + 2*zw, 0, DD-1);
    int gy = clampi(by + 2*yw, 0, HH-1);
    int gx = clampi(bx + 2*xw, 0, WW-1);
    const float* src = fmov + ((size_t)c*DHW + (size_t)gz*HWXY + (size_t)gy*WW + gx);
    float* dst = &s_mov[c][zw][yw][xw];
#if defined(HAVE_ASYNC_LDS)
    async_copy_b32(src, dst);
#else
    *dst = *src;
#endif
  }

  // ---- Stage fixed patch (regular loads; scaled by -2 for the WMMA path) ----
  #pragma unroll
  for (int l = 0; l < (CCH*PW*PW*PW)/NTHREADS; ++l) {      // 4 iters
    int i  = tid + l*NTHREADS;
    int px = i % PW;
    int t1 = i / PW;
    int py = t1 % PW;
    int t2 = t1 / PW;
    int pz = t2 % PW;
    int c  = t2 / PW;
    int gz = clampi(kz - 3 + 2*pz, 0, DD-1);
    int gy = clampi(ky - 3 + 2*py, 0, HH-1);
    int gx = clampi(kx - 3 + 2*px, 0, WW-1);
    float v = ffix[(size_t)c*DHW + (size_t)gz*HWXY + (size_t)gy*WW + gx];
#if defined(HAVE_WMMA_F32)
    v = -2.0f * v;
#endif
    (&s_pf[0][0][0][0])[i] = v;
  }

#if defined(HAVE_ASYNC_LDS)
  wait_async_zero();
#endif
  __syncthreads();

#if defined(HAVE_WMMA_F32)
  // ===================== WMMA path =====================
  // S_ff reduction: s_pf holds -2*pf, so sum of squares = 4*S_ff.
  {
    const float4 q = reinterpret_cast<const float4*>(&s_pf[0][0][0][0])[tid];
    float partial = q.x*q.x + q.y*q.y + q.z*q.z + q.w*q.w;
    atomicAdd(&s_sff, partial);
  }
  __syncthreads();

  // Per-lane fragment roles.
  const int wave  = tid >> 5;          // N tile 0..5
  const int lane  = tid & 31;
  const int half  = lane >> 4;         // 0/1
  const int li    = lane & 15;
  const int m     = li;                // dz row of A/D
  const bool mok  = (m < LW);          // valid dz rows 0..8
  const int n     = wave*16 + li;      // global column (dy*9+dx)
  const bool nok  = (n < LW*LW);
  const int nc    = nok ? n : (LW*LW - 1);
  const int dy    = nc / LW;
  const int dx    = nc % LW;
  const int p0    = half*2;            // K-pair within chunk: px = p0, p0+1

  v8f acc = {0.f,0.f,0.f,0.f,0.f,0.f,0.f,0.f};

  #pragma unroll 1
  for (int c = 0; c < CCH; ++c) {
    #pragma unroll 1
    for (int zw = 0; zw < WIN; ++zw) {
      const int pzr = zw - m;
      const bool band = mok && ((unsigned)pzr < (unsigned)PW);
      const int pzc = clampi(pzr, 0, PW-1);
      // Band mask applied by MULTIPLY (keeps LDS loads unconditional -> no
      // EXEC-mask predication blocks around ds_load).
      const float bandf = band ? 1.0f : 0.0f;
      const v2f A2 = {bandf, bandf};
      #pragma unroll
      for (int py = 0; py < PW; ++py) {
        // A fragment: banded -2*pf values (px = p0, p0+1), 8B-aligned pair.
        const v2f av = *reinterpret_cast<const v2f*>(&s_pf[c][pzc][py][p0]);
        const v2f A1 = {av.x * bandf, av.y * bandf};
        // B fragment: window values (px = p0, p0+1 shifted by dx)
        const float* bp = &s_mov[c][zw][py + dy][p0 + dx];
        const float b0 = bp[0];
        const float b1 = bp[1];
        const v2f B1 = {b0, b1};
        const v2f B2 = {b0*b0, b1*b1};
        acc = wmma_f32_4(A1, B1, acc);   // -2*pf*mov contributions
        acc = wmma_f32_4(A2, B2, acc);   // mov^2 contributions
      }
    }
  }

  // Epilogue: D[m,n] + S_ff, scaled by 1/768.  S_ff = s_sff/4.
  {
    const float sff = s_sff * 0.25f;
    if (nok) {
      float* ob = out + (size_t)k*NDISP + n;
      #pragma unroll
      for (int i = 0; i < 8; ++i) {
        const int mm = i + half*8;
        if (mm < LW) ob[(size_t)mm*(LW*LW)] = (acc[i] + sff) * (1.0f/768.0f);
      }
    }
  }
#else
  // ===================== VALU fallback path =====================
  if (tid < LW*LW) {
    const int dz = tid / LW;
    const int dy = tid % LW;
    float accv[LW];
    #pragma unroll
    for (int i = 0; i < LW; ++i) accv[i] = 0.0f;
    #pragma unroll 1
    for (int c = 0; c < CCH; ++c) {
      #pragma unroll 1
      for (int pz = 0; pz < PW; ++pz) {
        const int zw = pz + dz;
        #pragma unroll
        for (int py = 0; py < PW; ++py) {
          const int yw = py + dy;
          const float4* rp = reinterpret_cast<const float4*>(&s_mov[c][zw][yw][0]);
          const float4 r0 = rp[0], r1 = rp[1], r2 = rp[2];
          const float r[WIN] = { r0.x, r0.y, r0.z, r0.w,
                                 r1.x, r1.y, r1.z, r1.w,
                                 r2.x, r2.y, r2.z, r2.w };
          const float q0 = s_pf[c][pz][py][0];
          const float q1 = s_pf[c][pz][py][1];
          const float q2 = s_pf[c][pz][py][2];
          const float q3 = s_pf[c][pz][py][3];
          #pragma unroll
          for (int dx = 0; dx < LW; ++dx) {
            float d0 = q0 - r[dx + 0];
            float d1 = q1 - r[dx + 1];
            float d2 = q2 - r[dx + 2];
            float d3 = q3 - r[dx + 3];
            float a = accv[dx];
            a = fmaf(d0, d0, a);
            a = fmaf(d1, d1, a);
            a = fmaf(d2, d2, a);
            a = fmaf(d3, d3, a);
            accv[dx] = a;
          }
        }
      }
    }
    float* o = out + (size_t)k*NDISP + (size_t)(dz*LW + dy)*LW;
    #pragma unroll
    for (int dx = 0; dx < LW; ++dx) o[dx] = accv[dx] * (1.0f/768.0f);
  }
#endif

  // ---- Second output (disp grid, constant): written by block 0 ----
  if (k == 0) {
    for (int j = tid; j < NDISP*3; j += NTHREADS) {
      int d  = j / 3;
      int ax = j % 3;
      int comp = (ax == 0) ? (d / 81) : (ax == 1) ? ((d / 9) % 9) : (d % 9);
      out[COST_ELEMS + j] = (float)(-8 + 2*comp);
    }
  }
}

extern "C" void kernel_launch(void* const* d_in, const int* in_sizes, int n_in,
                              void* d_out, int out_size, void* d_ws, size_t ws_size,
                              hipStream_t stream) {
  const int*   kps  = (const int*)d_in[0];    // fix_kps (512,3) int32
  const float* ffix = (const float*)d_in[1];  // feat_fix (1,12,128,160,160) f32
  const float* fmov = (const float*)d_in[2];  // feat_mov f32
  float* out = (float*)d_out;                 // cost (512*729) ++ disp (729*3)

  hipLaunchKernelGGL(match_cost_kernel, dim3(NKPS), dim3(NTHREADS), 0, stream,
                     kps, ffix, fmov, out);
}